// HTop1Gate_57062935495438
// MI455X (gfx1250) — compile-verified
//
#include <hip/hip_runtime.h>
#include <hip/hip_bf16.h>

typedef float v2f __attribute__((ext_vector_type(2)));
typedef float v4f __attribute__((ext_vector_type(4)));
typedef float v8f __attribute__((ext_vector_type(8)));

#define S_TOK 4096
#define DDIM  1024
#define EEXP  64
#define CAPACITY 512
#define NEGV (-1e9f)
static const size_t COMB_ELEMS = (size_t)S_TOK * EEXP * CAPACITY; // 134,217,728

// ---------------------------------------------------------------------------
// Zero the (huge, mostly-sparse) output with non-temporal 16B stores.
// ---------------------------------------------------------------------------
__global__ void htop1_zero(float* __restrict__ out, size_t n) {
    size_t i = (size_t)blockIdx.x * blockDim.x + threadIdx.x;
    size_t stride = (size_t)gridDim.x * blockDim.x;
    size_t n4 = n >> 2;
    v4f* o4 = (v4f*)out;
    v4f z = {0.f, 0.f, 0.f, 0.f};
    for (size_t j = i; j < n4; j += stride)
        __builtin_nontemporal_store(z, &o4[j]);
    for (size_t j = (n4 << 2) + i; j < n; j += stride)
        out[j] = 0.f;
}

// ---------------------------------------------------------------------------
// GEMM: Out[S,E] = X[S,D] @ W[E,D]^T  via V_WMMA_F32_16X16X4_F32.
// One wave per 16-row stripe; A fragment reused across the 4 N-tiles (E=64).
// f32 A 16x4 layout: lanes 0-15 M=lane, K={k0,k0+1}; lanes 16-31 K={k0+2,k0+3}
//   -> one contiguous float2 per lane from row-major X.
// f32 B 4x16 layout (transpose-symmetric): lane n=lane&15, same K split
//   -> one contiguous float2 per lane from row-major W (B = W^T).
// ---------------------------------------------------------------------------
__global__ void htop1_gemm(const float* __restrict__ X, const float* __restrict__ W,
                           float* __restrict__ Out) {
    const int lane = threadIdx.x;       // 0..31
    const int tileM = blockIdx.x;       // 0..255
    const int half = lane >> 4;         // 0 or 1
    const int mr   = lane & 15;
    const int kb   = half * 2;

    const float* a_ptr = X + (size_t)(tileM * 16 + mr) * DDIM + kb;
    const float* b_ptr0 = W + (size_t)(mr +  0) * DDIM + kb;
    const float* b_ptr1 = W + (size_t)(mr + 16) * DDIM + kb;
    const float* b_ptr2 = W + (size_t)(mr + 32) * DDIM + kb;
    const float* b_ptr3 = W + (size_t)(mr + 48) * DDIM + kb;

    v8f acc0 = {}, acc1 = {}, acc2 = {}, acc3 = {};
    for (int k = 0; k < DDIM; k += 4) {
        float2 af = *(const float2*)(a_ptr + k);
        v2f a; a.x = af.x; a.y = af.y;
        float2 bf0 = *(const float2*)(b_ptr0 + k);
        float2 bf1 = *(const float2*)(b_ptr1 + k);
        float2 bf2 = *(const float2*)(b_ptr2 + k);
        float2 bf3 = *(const float2*)(b_ptr3 + k);
        v2f b0; b0.x = bf0.x; b0.y = bf0.y;
        v2f b1; b1.x = bf1.x; b1.y = bf1.y;
        v2f b2; b2.x = bf2.x; b2.y = bf2.y;
        v2f b3; b3.x = bf3.x; b3.y = bf3.y;
        acc0 = __builtin_amdgcn_wmma_f32_16x16x4_f32(false, a, false, b0, (short)0, acc0, false, false);
        acc1 = __builtin_amdgcn_wmma_f32_16x16x4_f32(false, a, false, b1, (short)0, acc1, false, false);
        acc2 = __builtin_amdgcn_wmma_f32_16x16x4_f32(false, a, false, b2, (short)0, acc2, false, false);
        acc3 = __builtin_amdgcn_wmma_f32_16x16x4_f32(false, a, false, b3, (short)0, acc3, false, false);
    }
    // C/D 16x16 f32 layout: VGPR r -> M = r + (lane<16 ? 0 : 8), N = lane&15
    const int mbase = tileM * 16 + half * 8;
#pragma unroll
    for (int r = 0; r < 8; ++r) {
        size_t rowoff = (size_t)(mbase + r) * EEXP + mr;
        Out[rowoff +  0] = acc0[r];
        Out[rowoff + 16] = acc1[r];
        Out[rowoff + 32] = acc2[r];
        Out[rowoff + 48] = acc3[r];
    }
}

// ---------------------------------------------------------------------------
// Per-token gating: one wave per token. Lane L holds experts L and L+32.
// top-16 group selection (monotonic in logits), masked softmax over the
// second-level logits, argmax -> (expert, gate value).
// ---------------------------------------------------------------------------
__device__ __forceinline__ float wave_max(float v) {
#pragma unroll
    for (int off = 16; off; off >>= 1) v = fmaxf(v, __shfl_xor(v, off, 32));
    return v;
}
__device__ __forceinline__ float wave_sum(float v) {
#pragma unroll
    for (int off = 16; off; off >>= 1) v += __shfl_xor(v, off, 32);
    return v;
}
__device__ __forceinline__ int wave_min_i(int v) {
#pragma unroll
    for (int off = 16; off; off >>= 1) v = min(v, __shfl_xor(v, off, 32));
    return v;
}

__global__ void htop1_gate(const float* __restrict__ glog, const float* __restrict__ slog,
                           float* __restrict__ gates, int* __restrict__ eidx,
                           float* __restrict__ gval) {
    const int lane = threadIdx.x & 31;
    const int wave = threadIdx.x >> 5;
    const int s = blockIdx.x * (blockDim.x >> 5) + wave;
    if (s >= S_TOK) return;

    // first-level group logits (col 0 masked to NEG before softmax/top-k)
    float v0 = glog[(size_t)s * EEXP + lane];
    float v1 = glog[(size_t)s * EEXP + lane + 32];
    if (lane == 0) v0 = NEGV;

    bool m0 = false, m1 = false;
    for (int it = 0; it < 16; ++it) {          // top-16 groups
        float mx = wave_max(fmaxf(v0, v1));
        int cand = (v0 == mx) ? lane : ((v1 == mx) ? (lane + 32) : 1000);
        int sel = wave_min_i(cand);            // first index on ties (top_k semantics)
        if (sel == lane)            { m0 = true; v0 = NEGV; }
        else if (sel == lane + 32)  { m1 = true; v1 = NEGV; }
    }

    // second-level masked softmax
    float l0 = m0 ? slog[(size_t)s * EEXP + lane]      : NEGV;
    float l1 = m1 ? slog[(size_t)s * EEXP + lane + 32] : NEGV;
    float mx = wave_max(fmaxf(l0, l1));
    float e0 = expf(l0 - mx);
    float e1 = expf(l1 - mx);
    float inv = 1.0f / wave_sum(e0 + e1);
    float g0 = e0 * inv, g1 = e1 * inv;
    gates[(size_t)s * EEXP + lane]      = g0;
    gates[(size_t)s * EEXP + lane + 32] = g1;

    // argmax of gates == first index achieving the masked-logit max
    int cand = (l0 == mx) ? lane : ((l1 == mx) ? (lane + 32) : 1000);
    int ei = wave_min_i(cand);
    float candg = (ei < 32) ? g0 : g1;
    float gv = __shfl(candg, ei & 31, 32);
    if (lane == 0) { eidx[s] = ei; gval[s] = gv; }
}

// ---------------------------------------------------------------------------
// Per-expert token-order scan (cumsum(mask1)) + l_aux.
// Single block, 64 threads (one per expert). Reads are block-broadcast.
// ---------------------------------------------------------------------------
__global__ void htop1_scan(const float* __restrict__ gates, const int* __restrict__ eidx,
                           int* __restrict__ loc, float* __restrict__ out_laux) {
    __shared__ float part[EEXP];
    const int e = threadIdx.x;   // 0..63
    float mesum = 0.f;
    int cnt = 0;
    for (int s = 0; s < S_TOK; ++s) {
        mesum += gates[(size_t)s * EEXP + e];
        if (eidx[s] == e) { loc[s] = cnt; ++cnt; }
    }
    float me = mesum * (1.0f / S_TOK);
    float ce = (float)cnt * (1.0f / S_TOK);
    part[e] = me * ce;
    __syncthreads();
    if (e == 0) {
        float t = 0.f;
        for (int i = 0; i < EEXP; ++i) t += part[i];
        out_laux[0] = t * 256.0f;   // sum/num_2nd*(E*E) = sum/16*4096
    }
}

// ---------------------------------------------------------------------------
// Scatter the <=4096 nonzeros of combine/dispatch (tokens past capacity drop).
// ---------------------------------------------------------------------------
__global__ void htop1_scatter(const int* __restrict__ eidx, const int* __restrict__ loc,
                              const float* __restrict__ gval, float* __restrict__ out) {
    const int s = blockIdx.x * blockDim.x + threadIdx.x;
    if (s >= S_TOK) return;
    const int l = loc[s];
    if (l < CAPACITY) {
        size_t ci = 1 + (((size_t)s * EEXP + eidx[s]) * CAPACITY + l);
        out[ci] = gval[s];                 // combine
        out[ci + COMB_ELEMS] = 1.0f;       // dispatch (bool -> 1.0f)
    }
}

// ---------------------------------------------------------------------------
extern "C" void kernel_launch(void* const* d_in, const int* in_sizes, int n_in,
                              void* d_out, int out_size, void* d_ws, size_t ws_size,
                              hipStream_t stream) {
    const float* input1 = (const float*)d_in[0];
    const float* input2 = (const float*)d_in[1];
    const float* wg1    = (const float*)d_in[2];
    const float* wg2    = (const float*)d_in[3];
    float* out = (float*)d_out;

    // workspace layout
    float* glog  = (float*)d_ws;                      // [S,E]
    float* slog  = glog + (size_t)S_TOK * EEXP;       // [S,E]
    float* gates = slog + (size_t)S_TOK * EEXP;       // [S,E]
    int*   eidx  = (int*)(gates + (size_t)S_TOK * EEXP); // [S]
    float* gval  = (float*)(eidx + S_TOK);            // [S]
    int*   loc   = (int*)(gval + S_TOK);              // [S]

    // 1) stream-zero the ~1.07 GB output (bandwidth-dominant step)
    htop1_zero<<<8192, 256, 0, stream>>>(out, (size_t)out_size);

    // 2) the two gating GEMMs (f32 WMMA)
    htop1_gemm<<<S_TOK / 16, 32, 0, stream>>>(input1, wg1, glog);
    htop1_gemm<<<S_TOK / 16, 32, 0, stream>>>(input2, wg2, slog);

    // 3) per-token gating (wave per token, 8 waves / block)
    htop1_gate<<<S_TOK / 8, 256, 0, stream>>>(glog, slog, gates, eidx, gval);

    // 4) per-expert token-order scan + l_aux
    htop1_scan<<<1, EEXP, 0, stream>>>(gates, eidx, loc, out);

    // 5) scatter nonzeros of combine / dispatch
    htop1_scatter<<<S_TOK / 256, 256, 0, stream>>>(eidx, loc, gval, out);
}